// DynamicConv2d_88227218194858
// MI455X (gfx1250) — compile-verified
//
#include <hip/hip_runtime.h>

typedef __attribute__((ext_vector_type(2))) float v2f;
typedef __attribute__((ext_vector_type(8))) float v8f;

#define B_N    16
#define CIN    64
#define COUT   64
#define H_IN   128
#define W_IN   128
#define HOUT   126
#define WOUT   126
#define KDIM   576                    /* C_in * 3 * 3 */

#define WPITCH 578                    /* sW row pitch: 8B-aligned lane bases, bank-spread */
#define XCOLS  130                    /* input cols needed: 128 px + 2 halo */
#define CIPITCH 66                    /* sX ci pitch: 8B-aligned, bank-spread */

#define SW_FLOATS (COUT * WPITCH)             /* 36992 */
#define SX_FLOATS (3 * XCOLS * CIPITCH)       /* 25740 */
#define LDS_BYTES ((SW_FLOATS + SX_FLOATS) * 4)  /* 250928 < 320K WGP LDS */

// Implicit GEMM per batch on the fp32 WMMA pipe, K reordered as k' = tap*64+ci
// so every fragment address = lane-constant base + compile-time ds offset.
// Block = (batch b, output row h, all 128 pixels). 8 waves = 4 co-tiles x 2
// pixel halves; each wave: 4 accumulators (16 co x 64 px), inner loop is
// 1x ds_load_b64 (A) + 4x ds_load_b64 (B) + 4x v_wmma_f32_16x16x4_f32.
extern "C" __global__ __launch_bounds__(256)
void dynconv2d_wmma_f32(const float* __restrict__ x,
                        const float* __restrict__ w,
                        const float* __restrict__ bias,
                        float* __restrict__ out)
{
    extern __shared__ float smem[];
    float* sW = smem;                 // [co][k' = r*64+ci], pitch WPITCH
    float* sX = smem + SW_FLOATS;     // [kh][col][ci],      ci pitch CIPITCH

    const int h   = blockIdx.x;       // 0..125
    const int b   = blockIdx.y;       // 0..15
    const int tid = threadIdx.x;

    // ---- Stage weights: transpose [ci][r] -> [r][ci] inside each co row ----
    const float* wsrc = w + (size_t)b * COUT * KDIM;
    for (int i = tid; i < COUT * KDIM; i += 256) {   // coalesced global read
        int co  = i / KDIM;
        int rem = i - co * KDIM;
        int ci  = rem / 9;
        int r   = rem - ci * 9;
        sW[co * WPITCH + r * 64 + ci] = wsrc[i];
    }

    // ---- Stage x rows h..h+2, transposed to [kh][col][ci] ----
    for (int i = tid; i < CIN * 3 * XCOLS; i += 256) {
        int ci  = i / (3 * XCOLS);
        int rem = i - ci * (3 * XCOLS);
        int kh  = rem / XCOLS;
        int col = rem - kh * XCOLS;
        float v = 0.0f;
        if (col < W_IN)
            v = x[(((size_t)b * CIN + ci) * H_IN + (h + kh)) * W_IN + col];
        sX[(kh * XCOLS + col) * CIPITCH + ci] = v;
    }
    __syncthreads();

    const int wave  = tid >> 5;
    const int lane  = tid & 31;
    const int laneN = lane & 15;      // A row (co) / B-D col (pixel)
    const int hi    = lane >> 4;      // K sub-pair: {0,1} vs {2,3}

    const int co_base = (wave & 3) * 16;
    const int P0      = (wave >> 2) * 64;   // pixel base: 0 or 64

    // Lane-constant LDS bases (all byte offsets are multiples of 8 -> b64 ok).
    const float* aB  = sW + (co_base + laneN) * WPITCH + 2 * hi;
    const float* bB0 = sX + (P0 +  0 + laneN) * CIPITCH + 2 * hi;
    const float* bB1 = sX + (P0 + 16 + laneN) * CIPITCH + 2 * hi;
    const float* bB2 = sX + (P0 + 32 + laneN) * CIPITCH + 2 * hi;
    const float* bB3 = sX + (P0 + 48 + laneN) * CIPITCH + 2 * hi;

    v8f acc0 = {}, acc1 = {}, acc2 = {}, acc3 = {};

    #pragma unroll
    for (int kh = 0; kh < 3; ++kh) {
        #pragma unroll
        for (int kw = 0; kw < 3; ++kw) {
            const int r    = kh * 3 + kw;
            const int aoff = r * 64;                         // floats
            const int boff = (kh * XCOLS + kw) * CIPITCH;    // floats
            #pragma unroll
            for (int c0 = 0; c0 < CIN; c0 += 4) {
                v2f a  = *(const v2f*)(aB  + aoff + c0);
                v2f b0 = *(const v2f*)(bB0 + boff + c0);
                v2f b1 = *(const v2f*)(bB1 + boff + c0);
                v2f b2 = *(const v2f*)(bB2 + boff + c0);
                v2f b3 = *(const v2f*)(bB3 + boff + c0);
                acc0 = __builtin_amdgcn_wmma_f32_16x16x4_f32(false, a, false, b0, (short)0, acc0, false, false);
                acc1 = __builtin_amdgcn_wmma_f32_16x16x4_f32(false, a, false, b1, (short)0, acc1, false, false);
                acc2 = __builtin_amdgcn_wmma_f32_16x16x4_f32(false, a, false, b2, (short)0, acc2, false, false);
                acc3 = __builtin_amdgcn_wmma_f32_16x16x4_f32(false, a, false, b3, (short)0, acc3, false, false);
            }
        }
    }

    // ---- Epilogue: bias + store. C/D layout: VGPR j -> M = j + 8*hi, N = laneN.
    #pragma unroll
    for (int t = 0; t < 4; ++t) {
        const v8f& acc = (t == 0) ? acc0 : (t == 1) ? acc1 : (t == 2) ? acc2 : acc3;
        const int wpix = P0 + t * 16 + laneN;
        if (wpix < WOUT) {
            #pragma unroll
            for (int j = 0; j < 8; ++j) {
                int co = co_base + j + 8 * hi;
                out[(((size_t)b * COUT + co) * HOUT + h) * WOUT + wpix] =
                    acc[j] + bias[b * COUT + co];
            }
        }
    }
}

extern "C" void kernel_launch(void* const* d_in, const int* in_sizes, int n_in,
                              void* d_out, int out_size, void* d_ws, size_t ws_size,
                              hipStream_t stream) {
    (void)in_sizes; (void)n_in; (void)out_size; (void)d_ws; (void)ws_size;
    const float* x    = (const float*)d_in[0];
    const float* w    = (const float*)d_in[1];
    const float* bias = (const float*)d_in[2];
    float* out        = (float*)d_out;

    dim3 grid(HOUT, B_N);   // (126, 16) blocks; one WGP-resident block each
    dynconv2d_wmma_f32<<<grid, 256, LDS_BYTES, stream>>>(x, w, bias, out);
}